// PTMAPLoss_50319836840464
// MI455X (gfx1250) — compile-verified
//
#include <hip/hip_runtime.h>

#define DD    4096
#define WAY   100
#define N_L   500
#define N_U   1500
#define N_TOT 2000
#define LAM   10.0f
#define ALPHA 0.2f
#define QUERYF 15.0f

typedef __attribute__((ext_vector_type(2))) float v2f;
typedef __attribute__((ext_vector_type(8))) float v8f;

// ---------------- preprocessing ----------------

__global__ void pow_norm_kernel(const float* __restrict__ X, float* __restrict__ Z) {
    int row = blockIdx.x;
    const float* x = X + (size_t)row * DD;
    float* z = Z + (size_t)row * DD;
    __shared__ float red[256];
    float s = 0.0f;
    for (int d = threadIdx.x; d < DD; d += 256) s += x[d] + 1e-6f;   // ||sqrt(x+eps)||^2
    red[threadIdx.x] = s; __syncthreads();
    for (int t = 128; t > 0; t >>= 1) { if (threadIdx.x < t) red[threadIdx.x] += red[threadIdx.x + t]; __syncthreads(); }
    float rn = 1.0f / fmaxf(sqrtf(red[0]), 1e-12f);
    for (int d = threadIdx.x; d < DD; d += 256) z[d] = sqrtf(x[d] + 1e-6f) * rn;
}

__global__ void col_means_kernel(const float* __restrict__ Z, float* __restrict__ ml, float* __restrict__ mu) {
    int d = blockIdx.x * blockDim.x + threadIdx.x;
    if (d >= DD) return;
    float s = 0.0f;
    for (int i = 0; i < N_L; ++i) s += Z[(size_t)i * DD + d];
    ml[d] = s * (1.0f / (float)N_L);
    s = 0.0f;
    for (int i = N_L; i < N_TOT; ++i) s += Z[(size_t)i * DD + d];
    mu[d] = s * (1.0f / (float)N_U);
}

__global__ void center_norm_kernel(float* __restrict__ Z, const float* __restrict__ ml, const float* __restrict__ mu) {
    int row = blockIdx.x;
    float* z = Z + (size_t)row * DD;
    const float* m = (row < N_L) ? ml : mu;
    __shared__ float red[256];
    float s = 0.0f;
    for (int d = threadIdx.x; d < DD; d += 256) { float v = z[d] - m[d]; s += v * v; }
    red[threadIdx.x] = s; __syncthreads();
    for (int t = 128; t > 0; t >>= 1) { if (threadIdx.x < t) red[threadIdx.x] += red[threadIdx.x + t]; __syncthreads(); }
    float rn = 1.0f / fmaxf(sqrtf(red[0]), 1e-12f);
    __syncthreads();
    for (int d = threadIdx.x; d < DD; d += 256) z[d] = (z[d] - m[d]) * rn;
}

__global__ void init_mus_kernel(const float* __restrict__ Z, float* __restrict__ Sl, float* __restrict__ mus) {
    int idx = blockIdx.x * 256 + threadIdx.x;          // < WAY*DD
    int c = idx >> 12, d = idx & (DD - 1);
    float s = 0.0f;
    for (int sh = 0; sh < 5; ++sh) s += Z[(size_t)(sh * WAY + c) * DD + d];
    Sl[idx] = s;
    mus[idx] = s * 0.2f;
}

__global__ void mus_norms_kernel(const float* __restrict__ mus, float* __restrict__ mn) {
    int c = blockIdx.x;
    const float* m = mus + (size_t)c * DD;
    __shared__ float red[256];
    float s = 0.0f;
    for (int d = threadIdx.x; d < DD; d += 256) s += m[d] * m[d];
    red[threadIdx.x] = s; __syncthreads();
    for (int t = 128; t > 0; t >>= 1) { if (threadIdx.x < t) red[threadIdx.x] += red[threadIdx.x + t]; __syncthreads(); }
    if (threadIdx.x == 0) mn[c] = red[0];
}

// ---------------- f32 WMMA GEMMs ----------------
// dist[i][c] = sqrt(max(1 + |mus_c|^2 - 2 * <Zu_i, mus_c>, 1e-12)), i<1500, c<100
// One wave computes one 16-row tile x TWO 16-class tiles (A-fragment reuse).
__global__ __launch_bounds__(32) void dist_wmma_kernel(const float* __restrict__ Z,
                                                       const float* __restrict__ mus,
                                                       const float* __restrict__ mn,
                                                       float* __restrict__ dist) {
    const float* Zu = Z + (size_t)N_L * DD;
    int i0 = blockIdx.x * 16;       // 94 tiles over 1500
    int cb = blockIdx.y * 32;       // 4 blocks x 2 tiles cover 0..127 >= 100
    int lane = threadIdx.x;
    int hl = lane >> 4;             // 0/1: which half of the wave
    int l15 = lane & 15;
    int kb = hl * 2;                // K sub-index for A/B VGPR pair
    int mrow = i0 + l15;            // A-matrix row (M) for this lane
    int nc0 = cb + l15;             // B-matrix col (N), tile 0
    int nc1 = cb + 16 + l15;        // B-matrix col (N), tile 1
    float amask = (mrow < N_U) ? 1.0f : 0.0f;
    float bm0 = (nc0 < WAY) ? 1.0f : 0.0f;
    float bm1 = (nc1 < WAY) ? 1.0f : 0.0f;
    const float* arow = Zu + (size_t)((mrow < N_U) ? mrow : 0) * DD;
    const float* br0 = mus + (size_t)((nc0 < WAY) ? nc0 : 0) * DD;
    const float* br1 = mus + (size_t)((nc1 < WAY) ? nc1 : 0) * DD;

    v8f acc0 = {}, acc1 = {};
    #pragma unroll 4
    for (int k = 0; k < DD; k += 4) {
        v2f a, b0, b1;
        a.x  = arow[k + kb]     * amask;
        a.y  = arow[k + kb + 1] * amask;
        b0.x = br0[k + kb]      * bm0;
        b0.y = br0[k + kb + 1]  * bm0;
        b1.x = br1[k + kb]      * bm1;
        b1.y = br1[k + kb + 1]  * bm1;
        acc0 = __builtin_amdgcn_wmma_f32_16x16x4_f32(false, a, false, b0, (short)0, acc0, false, false);
        acc1 = __builtin_amdgcn_wmma_f32_16x16x4_f32(false, a, false, b1, (short)0, acc1, false, false);
    }
    // C/D layout: lane -> column l15; VGPR v -> row v + 8*hl
    #pragma unroll
    for (int t = 0; t < 2; ++t) {
        int col = cb + 16 * t + l15;
        if (col < WAY) {
            float mnc = mn[col];
            #pragma unroll
            for (int v = 0; v < 8; ++v) {
                int row = i0 + v + 8 * hl;
                if (row < N_U) {
                    float g = (t == 0) ? acc0[v] : acc1[v];
                    float d2 = 1.0f + mnc - 2.0f * g;
                    dist[(size_t)row * WAY + col] = sqrtf(fmaxf(d2, 1e-12f));
                }
            }
        }
    }
}

// emus_raw[c][d] = Sl[c][d] + sum_i Pu[i][c] * Zu[i][d]
// One wave computes one 16-class tile x FOUR 16-wide d-tiles (reuses the Pu gathers).
__global__ __launch_bounds__(32) void emus_wmma_kernel(const float* __restrict__ Pu,
                                                       const float* __restrict__ Z,
                                                       const float* __restrict__ Sl,
                                                       float* __restrict__ emus) {
    const float* Zu = Z + (size_t)N_L * DD;
    int c0 = blockIdx.x * 16;       // 7 tiles over classes (M)
    int d0 = blockIdx.y * 64;       // 64 blocks x 4 tiles over feature dim (N)
    int lane = threadIdx.x;
    int hl = lane >> 4, l15 = lane & 15;
    int kb = hl * 2;
    int mcls = c0 + l15;
    float amask = (mcls < WAY) ? 1.0f : 0.0f;
    int acol = (mcls < WAY) ? mcls : 0;

    v8f acc[4] = {};
    #pragma unroll 2
    for (int k = 0; k < 1504; k += 4) {
        int k0 = k + kb, k1 = k + kb + 1;
        float m0 = (k0 < N_U) ? 1.0f : 0.0f;
        float m1 = (k1 < N_U) ? 1.0f : 0.0f;
        int kk0 = (k0 < N_U) ? k0 : 0;
        int kk1 = (k1 < N_U) ? k1 : 0;
        v2f a;
        a.x = Pu[(size_t)kk0 * WAY + acol] * (amask * m0);
        a.y = Pu[(size_t)kk1 * WAY + acol] * (amask * m1);
        const float* zr0 = Zu + (size_t)kk0 * DD + d0 + l15;
        const float* zr1 = Zu + (size_t)kk1 * DD + d0 + l15;
        #pragma unroll
        for (int t = 0; t < 4; ++t) {
            v2f b;
            b.x = zr0[16 * t];
            b.y = zr1[16 * t];
            acc[t] = __builtin_amdgcn_wmma_f32_16x16x4_f32(false, a, false, b, (short)0, acc[t], false, false);
        }
    }
    #pragma unroll
    for (int t = 0; t < 4; ++t) {
        int dcol = d0 + 16 * t + l15;
        #pragma unroll
        for (int v = 0; v < 8; ++v) {
            int crow = c0 + v + 8 * hl;
            if (crow < WAY)
                emus[(size_t)crow * DD + dcol] = Sl[(size_t)crow * DD + dcol] + acc[t][v];
        }
    }
}

__global__ void update_mus_kernel(float* __restrict__ mus, const float* __restrict__ emus,
                                  const float* __restrict__ psum) {
    int idx = blockIdx.x * 256 + threadIdx.x;          // < WAY*DD
    int c = idx >> 12;
    float e = emus[idx] / psum[c];
    float m = mus[idx];
    mus[idx] = m + ALPHA * (e - m);
}

// ---------------- persistent Sinkhorn (one 32-wave workgroup) ----------------
__global__ __launch_bounds__(1024) void sinkhorn_kernel(const float* __restrict__ dist,
                                                        float* __restrict__ Kb,
                                                        float* __restrict__ Pu,
                                                        float* __restrict__ psum) {
    __shared__ float a_s[1504];
    __shared__ float prev_s[1504];
    __shared__ float rs_s[1504];
    __shared__ float b_s[128];
    __shared__ float colpart[8][128];
    __shared__ float red[1024];
    __shared__ int done_s;
    const int tid = threadIdx.x;
    const int NE = N_U * WAY;       // 150000

    // K = exp(-LAM * M); K /= K.sum()
    float part = 0.0f;
    for (int idx = tid; idx < NE; idx += 1024) {
        float kv = __expf(-LAM * dist[idx]);
        Kb[idx] = kv; part += kv;
    }
    red[tid] = part; __syncthreads();
    for (int s = 512; s > 0; s >>= 1) { if (tid < s) red[tid] += red[tid + s]; __syncthreads(); }
    float inv = 1.0f / red[0];
    for (int idx = tid; idx < NE; idx += 1024) Kb[idx] *= inv;
    for (int i = tid; i < 1504; i += 1024) { a_s[i] = 1.0f; prev_s[i] = 0.0f; }
    if (tid < 128) b_s[tid] = 1.0f;
    if (tid == 0) done_s = 0;
    __syncthreads();

    for (int it = 0; it < 1000; ++it) {
        // rowsum = a * (K @ b)
        for (int i = tid; i < N_U; i += 1024) {
            const float* kr = Kb + (size_t)i * WAY;
            float dot = 0.0f;
            #pragma unroll 4
            for (int c = 0; c < WAY; ++c) dot += kr[c] * b_s[c];
            rs_s[i] = a_s[i] * dot;
        }
        __syncthreads();
        // err = max |prev - rowsum|
        float mx = 0.0f;
        for (int i = tid; i < N_U; i += 1024) mx = fmaxf(mx, fabsf(prev_s[i] - rs_s[i]));
        red[tid] = mx; __syncthreads();
        for (int s = 512; s > 0; s >>= 1) { if (tid < s) red[tid] = fmaxf(red[tid], red[tid + s]); __syncthreads(); }
        if (tid == 0 && red[0] <= 1e-6f) done_s = 1;
        __syncthreads();
        if (done_s) break;          // all remaining reference iterations are no-ops
        // a *= r / rowsum (r = 1); prev = rowsum
        for (int i = tid; i < N_U; i += 1024) { a_s[i] = a_s[i] / rs_s[i]; prev_s[i] = rs_s[i]; }
        __syncthreads();
        // colsum = b * (K^T @ a); b *= QUERY / colsum
        int c = tid & 127, g = tid >> 7;
        float acc = 0.0f;
        if (c < WAY) {
            for (int i = g; i < N_U; i += 8) {
                if (i + 64 < N_U)
                    __builtin_prefetch((const void*)(Kb + (size_t)(i + 64) * WAY + c), 0, 1);
                acc += Kb[(size_t)i * WAY + c] * a_s[i];
            }
        }
        colpart[g][c] = acc;
        __syncthreads();
        if (tid < WAY) {
            float cs = 0.0f;
            for (int gg = 0; gg < 8; ++gg) cs += colpart[gg][tid];
            cs *= b_s[tid];
            b_s[tid] = b_s[tid] * QUERYF / cs;
        }
        __syncthreads();
    }

    // P = a * K * b ; psum[c] = SHOT + sum_i P[i][c]
    int c = tid & 127, g = tid >> 7;
    float acc = 0.0f;
    if (c < WAY) {
        for (int i = g; i < N_U; i += 8) {
            float p = a_s[i] * Kb[(size_t)i * WAY + c] * b_s[c];
            Pu[(size_t)i * WAY + c] = p;
            acc += p;
        }
    }
    colpart[g][c] = acc;
    __syncthreads();
    if (tid < WAY) {
        float cs = 5.0f;
        for (int gg = 0; gg < 8; ++gg) cs += colpart[gg][tid];
        psum[tid] = cs;
    }
}

// ---------------- output ----------------
__global__ void zero_acc_kernel(float* acc) { if (blockIdx.x == 0 && threadIdx.x == 0) *acc = 0.0f; }

__global__ void finalize_kernel(const float* __restrict__ Pu, const int* __restrict__ labels,
                                float* __restrict__ out, float* __restrict__ acc) {
    int i = blockIdx.x;             // 1500 rows
    int tid = threadIdx.x;          // 128 threads
    __shared__ float bp[128];
    __shared__ int bi[128];
    float p = -1e30f; int best = tid;
    if (tid < WAY) {
        p = Pu[(size_t)i * WAY + tid];
        out[(size_t)i * WAY + tid] = logf(p + 1e-5f);
    }
    bp[tid] = p; bi[tid] = best; __syncthreads();
    for (int s = 64; s > 0; s >>= 1) {
        if (tid < s) {
            if (bp[tid + s] > bp[tid] || (bp[tid + s] == bp[tid] && bi[tid + s] < bi[tid])) {
                bp[tid] = bp[tid + s]; bi[tid] = bi[tid + s];
            }
        }
        __syncthreads();
    }
    if (tid == 0 && bi[0] == labels[N_L + i]) atomicAdd(acc, 1.0f);
}

__global__ void write_acc_kernel(const float* __restrict__ acc, float* __restrict__ out) {
    if (blockIdx.x == 0 && threadIdx.x == 0) out[(size_t)N_U * WAY] = *acc * (1.0f / (float)N_U);
}

// ---------------- host orchestration ----------------
extern "C" void kernel_launch(void* const* d_in, const int* in_sizes, int n_in,
                              void* d_out, int out_size, void* d_ws, size_t ws_size,
                              hipStream_t stream) {
    const float* X = (const float*)d_in[0];
    const int* labels = (const int*)d_in[1];
    float* out = (float*)d_out;

    float* ws = (float*)d_ws;
    size_t off = 0;
    float* Z    = ws + off; off += (size_t)N_TOT * DD;   // 8,192,000
    float* ml   = ws + off; off += DD;
    float* mu   = ws + off; off += DD;
    float* Sl   = ws + off; off += (size_t)WAY * DD;     // 409,600
    float* mus  = ws + off; off += (size_t)WAY * DD;
    float* emus = ws + off; off += (size_t)WAY * DD;
    float* mn   = ws + off; off += 128;
    float* dist = ws + off; off += (size_t)N_U * WAY;    // 150,000
    float* Kmat = ws + off; off += (size_t)N_U * WAY;
    float* Pu   = ws + off; off += (size_t)N_U * WAY;
    float* psum = ws + off; off += 128;
    float* accp = ws + off; off += 1;

    // preprocessing: power transform + normalize, center, renormalize, prototypes
    pow_norm_kernel<<<N_TOT, 256, 0, stream>>>(X, Z);
    col_means_kernel<<<DD / 256, 256, 0, stream>>>(Z, ml, mu);
    center_norm_kernel<<<N_TOT, 256, 0, stream>>>(Z, ml, mu);
    init_mus_kernel<<<(WAY * DD) / 256, 256, 0, stream>>>(Z, Sl, mus);

    dim3 dgrid((N_U + 15) / 16, 4);                      // 94 x 4 (each wave: 2 class tiles)
    dim3 egrid((WAY + 15) / 16, DD / 64);                // 7 x 64  (each wave: 4 d tiles)

    for (int e = 0; e < 21; ++e) {
        mus_norms_kernel<<<WAY, 256, 0, stream>>>(mus, mn);
        dist_wmma_kernel<<<dgrid, 32, 0, stream>>>(Z, mus, mn, dist);
        sinkhorn_kernel<<<1, 1024, 0, stream>>>(dist, Kmat, Pu, psum);
        if (e < 20) {
            emus_wmma_kernel<<<egrid, 32, 0, stream>>>(Pu, Z, Sl, emus);
            update_mus_kernel<<<(WAY * DD) / 256, 256, 0, stream>>>(mus, emus, psum);
        }
    }

    zero_acc_kernel<<<1, 1, 0, stream>>>(accp);
    finalize_kernel<<<N_U, 128, 0, stream>>>(Pu, labels, out, accp);
    write_acc_kernel<<<1, 1, 0, stream>>>(accp, out);
}